// AttentionDotProduct_77738908057604
// MI455X (gfx1250) — compile-verified
//
#include <hip/hip_runtime.h>

typedef __attribute__((ext_vector_type(2))) float v2f;
typedef __attribute__((ext_vector_type(8))) float v8f;

#define WAVES_PER_BLOCK 8

// One wave32 handles a 16-row query tile. Grid covers B*(S/16)=1024 tiles.
__global__ __launch_bounds__(256) void attn_causal_wmma_f32(
    const float* __restrict__ Q, const float* __restrict__ K,
    const float* __restrict__ V, float* __restrict__ outv,
    float* __restrict__ outw)
{
  constexpr int S = 4096, D = 64;
  // per-wave 16x16 staging tile for P (C-layout -> A-layout transpose);
  // row stride 18 floats (72B) keeps float2 loads 8B-aligned and avoids bank conflicts.
  __shared__ __align__(16) float ldsE[WAVES_PER_BLOCK][16][18];

  const int lane = threadIdx.x & 31;
  const int wid  = threadIdx.x >> 5;
  const int tile = blockIdx.x * WAVES_PER_BLOCK + wid;
  const int b    = tile >> 8;      // / (S/16)
  const int t    = tile & 255;     // q-tile index in batch
  const int q0   = t * 16;
  const int m    = lane & 15;      // N / M-row index within half
  const int h    = lane >> 4;      // which 16-lane half

  const float* Qb = Q + (size_t)b * S * D;
  const float* Kb = K + (size_t)b * S * D;
  const float* Vb = V + (size_t)b * S * D;
  float* Wb = outw + (size_t)b * S * S;
  float* Ob = outv + (size_t)b * S * D;

  // ---- zero-fill fully-masked (future) weight columns: softmax there is exactly 0
  {
    const int kz0 = q0 + 16;
    if (kz0 < S) {
      const int n4 = (S - kz0) >> 2;
      for (int r = 0; r < 16; ++r) {
        float4* p = (float4*)(Wb + (size_t)(q0 + r) * S + kz0);
        for (int c = lane; c < n4; c += 32)
          p[c] = make_float4(0.f, 0.f, 0.f, 0.f);
      }
    }
  }

  // ---- Q fragments (A-matrix layout, f32 16xK: lane m = row, vgpr pair = K 4c+2h, 4c+2h+1)
  v2f qf[16];
  {
    const float* qrow = Qb + (size_t)(q0 + m) * D + 2 * h;
#pragma unroll
    for (int c = 0; c < 16; ++c)
      qf[c] = *(const v2f*)(qrow + 4 * c);
  }

  float vmax[8];
#pragma unroll
  for (int r = 0; r < 8; ++r) vmax[r] = -3.0e38f;

  // ================= Pass 1: raw masked+scaled scores, running column max =================
  for (int j = 0; j <= t; ++j) {
    if (j < t)
      __builtin_prefetch(Kb + (size_t)(16 * (j + 1) + m) * D, 0, 3);
    // K fragments (B-matrix of Q*K^T): lane m = key row, vgpr pair = d 4c+2h, 4c+2h+1
    const float* krow = Kb + (size_t)(16 * j + m) * D + 2 * h;
    v2f kf[16];
#pragma unroll
    for (int c = 0; c < 16; ++c)
      kf[c] = *(const v2f*)(krow + 4 * c);

    v8f s8 = {};
#pragma unroll
    for (int c = 0; c < 16; ++c)
      s8 = __builtin_amdgcn_wmma_f32_16x16x4_f32(false, qf[c], false, kf[c],
                                                 (short)0, s8, false, false);

    const int kcol = 16 * j + m;
#pragma unroll
    for (int r = 0; r < 8; ++r) {
      const int qrow = q0 + r + 8 * h;       // C-layout: vgpr r -> row r (+8 for upper half)
      float s = s8[r];
      if (kcol > qrow) s -= 1.0e31f;         // mask BEFORE the sqrt(d) scale, like the reference
      s *= 0.125f;                           // 1/sqrt(64)
      Wb[(size_t)qrow * S + kcol] = s;       // lanes 0..15 -> contiguous 64B run per row
      vmax[r] = fmaxf(vmax[r], s);
    }
  }

  // row max: reduce across the 16 lanes of each half (masks < 16 stay in-half)
#pragma unroll
  for (int r = 0; r < 8; ++r) {
    float v = vmax[r];
    v = fmaxf(v, __shfl_xor(v, 1));
    v = fmaxf(v, __shfl_xor(v, 2));
    v = fmaxf(v, __shfl_xor(v, 4));
    v = fmaxf(v, __shfl_xor(v, 8));
    vmax[r] = v;
  }

  // ================= Pass 2: e = exp(s - m), row sums, P·V via WMMA =================
  float vsum[8];
#pragma unroll
  for (int r = 0; r < 8; ++r) vsum[r] = 0.f;
  v8f pv[4];
#pragma unroll
  for (int nt = 0; nt < 4; ++nt) pv[nt] = (v8f){};

  for (int j = 0; j <= t; ++j) {
    const int kcol = 16 * j + m;
#pragma unroll
    for (int r = 0; r < 8; ++r) {
      const int qrow = q0 + r + 8 * h;
      const float s  = Wb[(size_t)qrow * S + kcol];
      const float ev = __expf(s - vmax[r]);
      vsum[r] += ev;
      ldsE[wid][r + 8 * h][m] = ev;          // park e-tile for layout change
    }
    __builtin_amdgcn_wave_barrier();         // same-wave DS ops are in-order; just stop reordering

#pragma unroll
    for (int c = 0; c < 4; ++c) {
      // A fragment of P: lane m = query row, K pair = keys 4c+2h, 4c+2h+1
      const v2f a = *(const v2f*)&ldsE[wid][m][4 * c + 2 * h];
      const float* v0 = Vb + (size_t)(16 * j + 4 * c + 2 * h) * D + m;
      const float* v1 = v0 + D;
#pragma unroll
      for (int nt = 0; nt < 4; ++nt) {
        v2f bf;
        bf.x = v0[16 * nt];                  // B fragment of V: lane m = d column
        bf.y = v1[16 * nt];
        pv[nt] = __builtin_amdgcn_wmma_f32_16x16x4_f32(false, a, false, bf,
                                                       (short)0, pv[nt], false, false);
      }
    }
    __builtin_amdgcn_wave_barrier();         // ldsE reused next iteration
  }

  // row sums: same in-half reduction
#pragma unroll
  for (int r = 0; r < 8; ++r) {
    float v = vsum[r];
    v += __shfl_xor(v, 1);
    v += __shfl_xor(v, 2);
    v += __shfl_xor(v, 4);
    v += __shfl_xor(v, 8);
    vsum[r] = v;
  }

  float inv[8];
#pragma unroll
  for (int r = 0; r < 8; ++r) inv[r] = 1.0f / vsum[r];

  // ---- attn_vec = (P·V) / rowsum
#pragma unroll
  for (int nt = 0; nt < 4; ++nt)
#pragma unroll
    for (int r = 0; r < 8; ++r)
      Ob[(size_t)(q0 + r + 8 * h) * D + 16 * nt + m] = pv[nt][r] * inv[r];

  // ================= Pass 3: normalized weights = exp(s - m) / rowsum =================
  for (int j = 0; j <= t; ++j) {
    const int kcol = 16 * j + m;
#pragma unroll
    for (int r = 0; r < 8; ++r) {
      const size_t off = (size_t)(q0 + r + 8 * h) * S + kcol;
      const float s = Wb[off];
      Wb[off] = __expf(s - vmax[r]) * inv[r];
    }
  }
}

extern "C" void kernel_launch(void* const* d_in, const int* in_sizes, int n_in,
                              void* d_out, int out_size, void* d_ws, size_t ws_size,
                              hipStream_t stream) {
  (void)in_sizes; (void)n_in; (void)d_ws; (void)ws_size; (void)out_size;
  constexpr int B = 4, S = 4096, D = 64;
  const float* Q = (const float*)d_in[0];
  const float* K = (const float*)d_in[1];
  const float* V = (const float*)d_in[2];
  float* outv = (float*)d_out;                       // [B,S,D] first (return order)
  float* outw = outv + (size_t)B * S * D;            // then [B,S,S]
  const int tiles = B * (S / 16);                    // 1024
  dim3 grid(tiles / WAVES_PER_BLOCK);                // 128 blocks x 8 waves
  attn_causal_wmma_f32<<<grid, 256, 0, stream>>>(Q, K, V, outv, outw);
}